// SingleEmbedding_33157147525312
// MI455X (gfx1250) — compile-verified
//
#include <hip/hip_runtime.h>
#include <hip/hip_bf16.h>

#define N_ROWS 8192
#define DIM    768
#define TOPK   15
#define EPS    1e-8f

typedef __attribute__((ext_vector_type(2))) float v2f;
typedef __attribute__((ext_vector_type(8))) float v8f;

// ---------------------------------------------------------------------------
// Kernel 1: row-wise L2 normalize  W[8192x768] -> Wn (in workspace)
// ---------------------------------------------------------------------------
__global__ void __launch_bounds__(256)
normalize_rows(const float* __restrict__ W, float* __restrict__ Wn) {
    const int row = blockIdx.x;
    const float* wr = W  + (size_t)row * DIM;
    float*       wo = Wn + (size_t)row * DIM;

    float ss = 0.0f;
    for (int i = threadIdx.x; i < DIM; i += 256) {
        float v = wr[i];
        ss += v * v;
    }
    __shared__ float red[256];
    red[threadIdx.x] = ss;
    __syncthreads();
    for (int s = 128; s > 0; s >>= 1) {
        if (threadIdx.x < s) red[threadIdx.x] += red[threadIdx.x + s];
        __syncthreads();
    }
    const float norm  = sqrtf(red[0]);
    const float scale = 1.0f / fmaxf(norm, EPS);
    for (int i = threadIdx.x; i < DIM; i += 256) {
        wo[i] = wr[i] * scale;
    }
}

// ---------------------------------------------------------------------------
// Kernel 2: A = Wn * Wn^T via V_WMMA_F32_16X16X4_F32, register-blocked.
//
// Each wave owns a 64x64 macro-tile = 4x4 grid of 16x16 WMMA tiles
// (16 x v8f = 128 accumulator VGPRs).  Per k-step (K=4): load 4 A-fragments
// + 4 B-fragments (8 x global_load_b64) and issue 16 independent WMMAs
// -> 0.5 loads per WMMA (was 2.0), so the loop is WMMA-issue bound, not
// VMEM bound.  All fragment traffic hits L2 (Wn = 25 MB << 192 MB L2).
//
// Fragment layout (ISA 7.12.2, 32-bit A 16x4): lane = half*16 + m,
//   A vgpr0/1 = A[m][k + 2*half + {0,1}]   (row m of Wn)
//   B vgpr0/1 = B[k + 2*half + {0,1}][n]   = Wn[col n][k + 2*half + {0,1}]
// C/D layout: vgpr v, lane = half*16 + n  ->  D[v + 8*half][n].
//
// Block = 8 waves arranged 2x4 -> block tile 128 x 256.  Grid = (64, 32).
// ---------------------------------------------------------------------------
__global__ void __launch_bounds__(256)
cosine_gemm(const float* __restrict__ Wn, float* __restrict__ A) {
    const int wave = threadIdx.x >> 5;
    const int lane = threadIdx.x & 31;
    const int half = lane >> 4;    // 0 or 1
    const int lm   = lane & 15;

    const int rowBase = blockIdx.x * 128 + (wave >> 2) * 64;
    const int colBase = blockIdx.y * 256 + (wave & 3) * 64;

    const float* abase = Wn + (size_t)(rowBase + lm) * DIM + 2 * half;
    const float* bbase = Wn + (size_t)(colBase + lm) * DIM + 2 * half;

    v8f c[4][4] = {};

    #pragma unroll 2
    for (int k = 0; k < DIM; k += 4) {
        v2f a[4], b[4];
        #pragma unroll
        for (int r = 0; r < 4; ++r)
            a[r] = *(const v2f*)(abase + (size_t)(16 * r) * DIM + k);
        #pragma unroll
        for (int cc = 0; cc < 4; ++cc)
            b[cc] = *(const v2f*)(bbase + (size_t)(16 * cc) * DIM + k);

        #pragma unroll
        for (int r = 0; r < 4; ++r) {
            #pragma unroll
            for (int cc = 0; cc < 4; ++cc) {
                c[r][cc] = __builtin_amdgcn_wmma_f32_16x16x4_f32(
                        /*neg_a=*/false, a[r], /*neg_b=*/false, b[cc],
                        /*c_mod=*/(short)0, c[r][cc],
                        /*reuse_a=*/false, /*reuse_b=*/false);
            }
        }
    }

    #pragma unroll
    for (int r = 0; r < 4; ++r) {
        #pragma unroll
        for (int cc = 0; cc < 4; ++cc) {
            float* out = A + (size_t)(rowBase + 16 * r + half * 8) * N_ROWS
                           + (colBase + 16 * cc + lm);
            #pragma unroll
            for (int v = 0; v < 8; ++v) {
                out[(size_t)v * N_ROWS] = c[r][cc][v];
            }
        }
    }
}

// ---------------------------------------------------------------------------
// Kernel 3: per-row clamp(>=0), zero diagonal, top-15 (stable, lowest index
// wins ties like lax.top_k), masked rewrite of the row, and edge outputs.
// One 256-thread block per row; the 32KB row lives in LDS.
// Final outputs use non-temporal stores (never re-read; keep L2 for dense A).
// ---------------------------------------------------------------------------
__global__ void __launch_bounds__(256)
topk_row(float* __restrict__ A,
         float* __restrict__ e_src, float* __restrict__ e_dst,
         float* __restrict__ e_attr) {
    const int row = blockIdx.x;
    const int tid = threadIdx.x;

    __shared__ float rowv[N_ROWS];   // 32 KB
    __shared__ float rmax[256];
    __shared__ int   rmaxi[256];
    __shared__ float selV[TOPK];
    __shared__ int   selI[TOPK];

    float* arow = A + (size_t)row * N_ROWS;

    // Stage row with clamp + diagonal zero.
    for (int i = tid; i < N_ROWS; i += 256) {
        float v = fmaxf(arow[i], 0.0f);
        if (i == row) v = 0.0f;
        rowv[i] = v;
    }
    __syncthreads();

    // 15 rounds of block-wide argmax; winners get a negative sentinel
    // (-(val+1)) so they are excluded from later rounds (valid values >= 0).
    for (int t = 0; t < TOPK; ++t) {
        float bm = -1.0f;
        int   bi = N_ROWS;
        for (int i = tid; i < N_ROWS; i += 256) {
            float v = rowv[i];
            if (v > bm || (v == bm && i < bi)) { bm = v; bi = i; }
        }
        rmax[tid]  = bm;
        rmaxi[tid] = bi;
        __syncthreads();
        for (int s = 128; s > 0; s >>= 1) {
            if (tid < s) {
                float ov = rmax[tid + s];
                int   oi = rmaxi[tid + s];
                if (ov > rmax[tid] || (ov == rmax[tid] && oi < rmaxi[tid])) {
                    rmax[tid] = ov; rmaxi[tid] = oi;
                }
            }
            __syncthreads();
        }
        if (tid == 0) {
            selV[t] = rmax[0];
            selI[t] = rmaxi[0];
            rowv[rmaxi[0]] = -(rmax[0] + 1.0f);   // mark as selected
        }
        __syncthreads();
    }

    // Masked rewrite: selected positions keep their value, rest -> 0.
    // Non-temporal: terminal output, never re-read.
    for (int i = tid; i < N_ROWS; i += 256) {
        float v = rowv[i];
        __builtin_nontemporal_store((v < 0.0f) ? (-v - 1.0f) : 0.0f, &arow[i]);
    }

    // Edge list (indices stored as float, exact up to 2^24).
    if (tid < TOPK) {
        const size_t o = (size_t)row * TOPK + tid;
        __builtin_nontemporal_store((float)row,      &e_src[o]);
        __builtin_nontemporal_store((float)selI[tid], &e_dst[o]);
        __builtin_nontemporal_store(selV[tid],        &e_attr[o]);
    }
}

// ---------------------------------------------------------------------------
extern "C" void kernel_launch(void* const* d_in, const int* in_sizes, int n_in,
                              void* d_out, int out_size, void* d_ws, size_t ws_size,
                              hipStream_t stream) {
    const float* W = (const float*)d_in[0];
    // d_in[1] is the topk scalar (==15); baked in at compile time.

    float* out    = (float*)d_out;
    float* e_src  = out;                                  // [N*TOPK]
    float* e_dst  = out + (size_t)N_ROWS * TOPK;          // [N*TOPK]
    float* e_attr = out + (size_t)2 * N_ROWS * TOPK;      // [N*TOPK]
    float* A      = out + (size_t)3 * N_ROWS * TOPK;      // [N*N] dense/sparse

    float* Wn = (float*)d_ws;                             // 25 MB scratch

    normalize_rows<<<N_ROWS, 256, 0, stream>>>(W, Wn);

    dim3 g(N_ROWS / 128, N_ROWS / 256);
    cosine_gemm<<<g, 256, 0, stream>>>(Wn, A);

    topk_row<<<N_ROWS, 256, 0, stream>>>(A, e_src, e_dst, e_attr);
}